// gene_program_model_gcn_nonneg_22651657519233
// MI455X (gfx1250) — compile-verified
//
#include <hip/hip_runtime.h>

// ---------------- problem constants (from reference) ----------------
#define N_NODES 100000
#define N_EDGES 1600000
#define IN_DIM  128
#define HID     64
#define MLP_HID 256
#define OUT_DIM 128

typedef float v2f __attribute__((ext_vector_type(2)));
typedef float v8f __attribute__((ext_vector_type(8)));

// ---------------- degree / norm kernels ----------------
__global__ void __launch_bounds__(256) deg_init_kernel(float* deg, int n) {
    int i = blockIdx.x * 256 + threadIdx.x;
    if (i < n) deg[i] = 1.0f;                 // self-loop contributes 1
}

__global__ void __launch_bounds__(256) deg_count_kernel(const long long* __restrict__ ei,
                                                        float* __restrict__ deg, int nE) {
    int e = blockIdx.x * 256 + threadIdx.x;
    if (e < nE) {
        int dst = (int)ei[(size_t)N_EDGES + e];
        unsafeAtomicAdd(&deg[dst], 1.0f);     // global_atomic_add_f32
    }
}

__global__ void __launch_bounds__(256) deg_rsqrt_kernel(float* deg, int n) {
    int i = blockIdx.x * 256 + threadIdx.x;
    if (i < n) deg[i] = rsqrtf(deg[i]);       // deg > 0 always (self loops)
}

// ---------------- WMMA fp32 GEMM: C = [relu](relu?(A) @ B + bias?) ----------------
// A: MxK row-major, B: KxN row-major, C: MxN row-major. M%16==0, N%16==0, K%4==0.
// K, N, and all epilogue behavior are compile-time: inner loop is pure
// immediate-offset loads + V_WMMA_F32_16X16X4_F32.
// One wave per 16x16 C tile, 8 waves per block.
template <int K, int N, bool RELUA, bool BIAS, bool RELUOUT>
__global__ void __launch_bounds__(256) wmma_gemm_f32_kernel(
    const float* __restrict__ A, const float* __restrict__ B,
    const float* __restrict__ bias, float* __restrict__ C, int M)
{
    const int lane  = threadIdx.x & 31;
    const int wave  = threadIdx.x >> 5;
    const int mtile = blockIdx.x * 8 + wave;      // 16-row tile index
    if (mtile >= (M >> 4)) return;                // uniform per wave -> EXEC all-1s inside
    const int ntile = blockIdx.y;

    const int mrow = (mtile << 4) + (lane & 15);  // A row handled by this lane
    const int ncol = (ntile << 4) + (lane & 15);  // B/C column handled by this lane
    const int kb   = (lane >> 4) << 1;            // K sub-offset: lanes 0-15 -> K+0/1, 16-31 -> K+2/3

    // Per-lane base pointers; inner-loop offsets are compile-time immediates.
    const float* __restrict__ a0 = A + (size_t)mrow * K + kb;   // {a.x,a.y} = a0[k], a0[k+1]
    const float* __restrict__ b0 = B + (size_t)kb * N + ncol;   // b.x = b0[k*N]
    const float* __restrict__ b1 = b0 + N;                      // b.y = b1[k*N]

    v8f acc = {};                                 // 16x16 f32 accumulator (8 VGPRs)
#pragma unroll 8
    for (int k = 0; k < K; k += 4) {
        v2f a = *(const v2f*)(a0 + k);            // 8B-aligned global_load_b64
        if (RELUA) { a.x = fmaxf(a.x, 0.0f); a.y = fmaxf(a.y, 0.0f); }
        v2f b;
        b.x = b0[(size_t)k * N];
        b.y = b1[(size_t)k * N];
        acc = __builtin_amdgcn_wmma_f32_16x16x4_f32(
            /*neg_a=*/false, a, /*neg_b=*/false, b,
            /*c_mod=*/(short)0, acc, /*reuse_a=*/false, /*reuse_b=*/false);
    }

    // D layout: VGPR r holds M = r (lanes 0-15) / r+8 (lanes 16-31), N = lane&15
    const int mbase = (mtile << 4) + ((lane >> 4) << 3);
    const float bv = BIAS ? bias[ncol] : 0.0f;
    float* __restrict__ crow = C + (size_t)mbase * N + ncol;
#pragma unroll
    for (int r = 0; r < 8; ++r) {
        float v = acc[r];
        if (BIAS)    v += bv;
        if (RELUOUT) v = fmaxf(v, 0.0f);
        crow[(size_t)r * N] = v;
    }
}

// ---------------- GCN aggregation ----------------
// agg[i,f] = t[i,f] * dinv[i]^2 + b[f]   (self-loop term + bias; fully initializes agg)
__global__ void __launch_bounds__(256) agg_init_kernel(const float* __restrict__ t,
                                                       const float* __restrict__ dinv,
                                                       const float* __restrict__ b,
                                                       float* __restrict__ agg, int n) {
    int i = blockIdx.x * 256 + threadIdx.x;
    if (i < n) {
        int node = i >> 6;                // /HID (64)
        int f    = i & 63;
        float d  = dinv[node];
        agg[i] = t[i] * d * d + b[f];
    }
}

// agg[dst] += t[src] * dinv[src]*dinv[dst]; one thread per (edge, 4-feature group)
__global__ void __launch_bounds__(256) edge_scatter_kernel(const long long* __restrict__ ei,
                                                           const float* __restrict__ t,
                                                           const float* __restrict__ dinv,
                                                           float* __restrict__ agg) {
    int idx = blockIdx.x * 256 + threadIdx.x;     // N_EDGES*16 work items (fits in int)
    if (idx >= N_EDGES * 16) return;
    int e = idx >> 4;
    int g = (idx & 15) << 2;                      // feature base (groups of 4)
    int src = (int)ei[e];
    int dst = (int)ei[(size_t)N_EDGES + e];
    float norm = dinv[src] * dinv[dst];
    const float4 tv = *(const float4*)(t + (size_t)src * HID + g);  // 16B aligned
    float* ad = agg + (size_t)dst * HID + g;
    unsafeAtomicAdd(&ad[0], tv.x * norm);
    unsafeAtomicAdd(&ad[1], tv.y * norm);
    unsafeAtomicAdd(&ad[2], tv.z * norm);
    unsafeAtomicAdd(&ad[3], tv.w * norm);
}

// ---------------- host-side helpers ----------------
static inline dim3 gemm_grid(int M, int N) {
    return dim3((unsigned)((M / 16 + 7) / 8), (unsigned)(N / 16));
}

extern "C" void kernel_launch(void* const* d_in, const int* in_sizes, int n_in,
                              void* d_out, int out_size, void* d_ws, size_t ws_size,
                              hipStream_t stream) {
    const float*     x   = (const float*)d_in[0];
    const long long* ei  = (const long long*)d_in[1];   // int64 [2, E]
    const float*     W1  = (const float*)d_in[2];
    const float*     b1  = (const float*)d_in[3];
    const float*     W2  = (const float*)d_in[4];
    const float*     b2  = (const float*)d_in[5];
    const float*     Wf1 = (const float*)d_in[6];
    const float*     bf1 = (const float*)d_in[7];
    const float*     Wf2 = (const float*)d_in[8];
    const float*     bf2 = (const float*)d_in[9];
    const float*     Wf3 = (const float*)d_in[10];
    const float*     bf3 = (const float*)d_in[11];
    float* out = (float*)d_out;

    // workspace layout (256B aligned slabs)
    char*  ws  = (char*)d_ws;
    size_t off = 0;
    auto alloc = [&](size_t bytes) -> float* {
        float* p = (float*)(ws + off);
        off += (bytes + 255) & ~(size_t)255;
        return p;
    };
    float* dinv = alloc((size_t)N_NODES * sizeof(float));
    float* bufA = alloc((size_t)N_NODES * HID * sizeof(float));   // t1 / t2
    float* bufB = alloc((size_t)N_NODES * HID * sizeof(float));   // agg1 / agg2

    // MLP chunk buffers from remaining ws (deterministic: ws_size fixed per run)
    size_t remain  = (ws_size > off) ? (ws_size - off) : 0;
    size_t per_buf = remain / 2;
    long long chunk_ll = (long long)((per_buf / (MLP_HID * sizeof(float))) & ~(size_t)15);
    int chunk = (chunk_ll > N_NODES) ? N_NODES : (int)chunk_ll;
    if (chunk < 16) chunk = 16;
    float* bufC = alloc((size_t)chunk * MLP_HID * sizeof(float)); // h3 chunk
    float* bufD = alloc((size_t)chunk * MLP_HID * sizeof(float)); // h4 chunk

    // ---- degree / symmetric norm ----
    deg_init_kernel <<<(N_NODES + 255) / 256, 256, 0, stream>>>(dinv, N_NODES);
    deg_count_kernel<<<(N_EDGES + 255) / 256, 256, 0, stream>>>(ei, dinv, N_EDGES);
    deg_rsqrt_kernel<<<(N_NODES + 255) / 256, 256, 0, stream>>>(dinv, N_NODES);

    const int aggN   = N_NODES * HID;
    const int edgeWk = N_EDGES * 16;

    // ---- GCN layer 1: agg1 = norm-agg(x@W1) + b1  (ReLU fused into next GEMM A-read) ----
    wmma_gemm_f32_kernel<IN_DIM, HID, false, false, false>
        <<<gemm_grid(N_NODES, HID), 256, 0, stream>>>(x, W1, nullptr, bufA, N_NODES);
    agg_init_kernel    <<<(aggN   + 255) / 256, 256, 0, stream>>>(bufA, dinv, b1, bufB, aggN);
    edge_scatter_kernel<<<(edgeWk + 255) / 256, 256, 0, stream>>>(ei, bufA, dinv, bufB);

    // ---- GCN layer 2: t2 = relu(agg1)@W2 ; agg2 = norm-agg(t2) + b2 ----
    wmma_gemm_f32_kernel<HID, HID, true, false, false>
        <<<gemm_grid(N_NODES, HID), 256, 0, stream>>>(bufB, W2, nullptr, bufA, N_NODES);
    agg_init_kernel    <<<(aggN   + 255) / 256, 256, 0, stream>>>(bufA, dinv, b2, bufB, aggN);
    edge_scatter_kernel<<<(edgeWk + 255) / 256, 256, 0, stream>>>(ei, bufA, dinv, bufB);

    // ---- MLP (chunked over rows; 100000 and chunk are multiples of 16) ----
    for (int r0 = 0; r0 < N_NODES; r0 += chunk) {
        int m = N_NODES - r0; if (m > chunk) m = chunk;
        wmma_gemm_f32_kernel<HID, MLP_HID, true, true, true>           // h3 = relu(relu(agg2)@Wf1+bf1)
            <<<gemm_grid(m, MLP_HID), 256, 0, stream>>>(bufB + (size_t)r0 * HID, Wf1, bf1, bufC, m);
        wmma_gemm_f32_kernel<MLP_HID, MLP_HID, false, true, true>      // h4 = relu(h3@Wf2+bf2)
            <<<gemm_grid(m, MLP_HID), 256, 0, stream>>>(bufC, Wf2, bf2, bufD, m);
        wmma_gemm_f32_kernel<MLP_HID, OUT_DIM, false, true, true>      // out = relu(h4@Wf3+bf3)
            <<<gemm_grid(m, OUT_DIM), 256, 0, stream>>>(bufD, Wf3, bf3, out + (size_t)r0 * OUT_DIM, m);
    }
}